// maskNRIConv_42666205118897
// MI455X (gfx1250) — compile-verified
//
#include <hip/hip_runtime.h>

// ---------------------------------------------------------------------------
// maskNRIConv forward for MI455X (gfx1250, wave32, WMMA).
//
// Roofline reasoning: after factoring layer-1 through the one-hot incidence
// matrices, the only large compute is the per-edge second MLP layer:
//   32 batches * 4032 edges * 2 types * (2*256*256) = 33.8 GFLOP,
// while required HBM traffic is only ~35 MB (inputs, rel_type, small per-node
// intermediates, output). At 23.3 TB/s that is ~1.5 us of memory time, so the
// kernel must be organized to be matrix-core bound, not bandwidth bound.
// We therefore fuse layer2 + rel_type weighting + receiver-scatter into one
// kernel (one workgroup per (batch, receiver)), with the shared 64x256
// activation tile staged in LDS in exact V_WMMA_F32_16X16X4_F32 A-fragment
// order (64 KB, conflict-free ds_load_b64) and weights pre-packed into
// B-fragment order (coalesced global_load_b64, resident in the 192 MB L2).
// All GEMMs use the native fp32 WMMA (no precision loss vs. the fp32 ref).
// ---------------------------------------------------------------------------

typedef __attribute__((ext_vector_type(2))) float v2f;
typedef __attribute__((ext_vector_type(8))) float v8f;

#define NB     64          // N_BALL
#define BT     32          // BATCH
#define CH     256         // IN_CH == OUT_CH
#define ROWS   2048        // BT*NB
#define PKSZ   32768       // v2f elements per packed 256x256 weight matrix

__device__ __forceinline__ unsigned f2key(float f) {
    unsigned u = __float_as_uint(f);
    return (u & 0x80000000u) ? ~u : (u | 0x80000000u);   // order-preserving
}

// ---------------------------------------------------------------------------
// Exact median-rank threshold per score matrix: 4-pass radix select (MSB->LSB)
// over monotone uint keys. One workgroup per matrix. Keep = key >= thr.
// ---------------------------------------------------------------------------
__global__ void __launch_bounds__(256)
select_thresh_kernel(const float* __restrict__ s1, const float* __restrict__ s2,
                     const float* __restrict__ so1, const float* __restrict__ so2,
                     unsigned* __restrict__ thr) {
    __shared__ unsigned hist[256];
    __shared__ unsigned sh_prefix, sh_want;
    const int t = blockIdx.x;
    const float* p; unsigned n;
    switch (t) {
        case 0: p = s1;          n = 131072u; break;
        case 1: p = s1 + 131072; n = 131072u; break;
        case 2: p = s2;          n = 65536u;  break;
        case 3: p = s2 + 65536;  n = 65536u;  break;
        case 4: p = so1;         n = 131072u; break;
        default:p = so2;         n = 65536u;  break;
    }
    if (threadIdx.x == 0) { sh_prefix = 0u; sh_want = n >> 1; }  // rank j = size/2
    for (int shift = 24; shift >= 0; shift -= 8) {
        __syncthreads();
        hist[threadIdx.x] = 0u;            // blockDim == 256
        __syncthreads();
        unsigned prefix = sh_prefix;
        for (unsigned i = threadIdx.x; i < n; i += 256u) {
            unsigned k = f2key(p[i]);
            bool match = (shift == 24) || ((k >> (shift + 8)) == prefix);
            if (match) atomicAdd(&hist[(k >> shift) & 255u], 1u);
        }
        __syncthreads();
        if (threadIdx.x == 0) {
            unsigned want = sh_want, cum = 0u, d = 0u;
            for (d = 0u; d < 256u; ++d) {
                unsigned h = hist[d];
                if (cum + h > want) break;
                cum += h;
            }
            if (d == 256u) d = 255u;
            sh_want = want - cum;
            sh_prefix = (sh_prefix << 8) | d;
        }
    }
    __syncthreads();
    if (threadIdx.x == 0) thr[t] = sh_prefix;
}

// ---------------------------------------------------------------------------
// Apply top-k mask and pack weights into B-fragment layout for
// V_WMMA_F32_16X16X4_F32: packed[(kt*16+nt)*32+lane] = float2 of
// W[nt*16 + lane%16, kt*4 + 2*(lane/16) + {0,1}]. 9 matrices, all 256x256.
// ---------------------------------------------------------------------------
__global__ void __launch_bounds__(256)
pack_weights_kernel(const float* __restrict__ w1, const float* __restrict__ s1,
                    const float* __restrict__ w2, const float* __restrict__ s2,
                    const float* __restrict__ wo1, const float* __restrict__ so1,
                    const float* __restrict__ wo2, const float* __restrict__ so2,
                    const unsigned* __restrict__ thr, v2f* __restrict__ pk) {
    const int which = blockIdx.y;
    const float *w, *s; int ldw, coff, ti;
    switch (which) {
        case 0: w = w1;          s = s1;          ldw = 512; coff = 0;   ti = 0; break;
        case 1: w = w1;          s = s1;          ldw = 512; coff = 256; ti = 0; break;
        case 2: w = w1 + 131072; s = s1 + 131072; ldw = 512; coff = 0;   ti = 1; break;
        case 3: w = w1 + 131072; s = s1 + 131072; ldw = 512; coff = 256; ti = 1; break;
        case 4: w = w2;          s = s2;          ldw = 256; coff = 0;   ti = 2; break;
        case 5: w = w2 + 65536;  s = s2 + 65536;  ldw = 256; coff = 0;   ti = 3; break;
        case 6: w = wo1;         s = so1;         ldw = 512; coff = 0;   ti = 4; break;
        case 7: w = wo1;         s = so1;         ldw = 512; coff = 256; ti = 4; break;
        default:w = wo2;         s = so2;         ldw = 256; coff = 0;   ti = 5; break;
    }
    const unsigned T = thr[ti];
    const int pid  = blockIdx.x * 256 + threadIdx.x;   // 0..32767
    const int lane = pid & 31;
    const int nt   = (pid >> 5) & 15;
    const int kt   = pid >> 9;
    const int row  = nt * 16 + (lane & 15);
    const int col  = kt * 4 + 2 * (lane >> 4);
    const int base = row * ldw + coff + col;
    v2f o;
    o.x = (f2key(s[base])     >= T) ? w[base]     : 0.f;
    o.y = (f2key(s[base + 1]) >= T) ? w[base + 1] : 0.f;
    pk[(size_t)which * PKSZ + pid] = o;
}

// ---------------------------------------------------------------------------
// rtw[b][r][i][s] = rel_type[b, edge(r,s), i], 0 on the diagonal (s==r).
// ---------------------------------------------------------------------------
__global__ void __launch_bounds__(256)
build_rtw_kernel(const float* __restrict__ rel_type, float* __restrict__ rtw) {
    const int idx = blockIdx.x * 256 + threadIdx.x;    // 0..262143
    const int s = idx & 63;
    const int i = (idx >> 6) & 1;
    const int r = (idx >> 7) & 63;
    const int b = idx >> 13;
    float v = 0.f;
    if (s != r) {
        int e = r * 63 + (s < r ? s : s - 1);
        v = rel_type[((size_t)b * 4032 + e) * 2 + i];
    }
    rtw[idx] = v;
}

// ---------------------------------------------------------------------------
// Generic fp32-WMMA GEMM: C[m,n] = act( sum_k A[m,k]*W[n,k] [+Cinit] ) [+Resid]
// M = grid.x*64 rows, K = N = 256. 256 threads = 8 waves; wave w owns a
// 32-column strip (two 16x16 N tiles) over four 16-row M tiles.
// A is staged in LDS in packed A-fragment order (64 KB, bank-conflict free).
// ---------------------------------------------------------------------------
__global__ void __launch_bounds__(256)
gemm256_wmma_kernel(const float* __restrict__ A, const v2f* __restrict__ Wp,
                    const float* __restrict__ Cinit, const float* __restrict__ Resid,
                    float* __restrict__ C, int doRelu) {
    __shared__ float As[16384];                       // 64 rows x 256 K, packed
    const int tid = threadIdx.x;
    const int m0  = blockIdx.x * 64;
    {   // stage: thread `tid` owns column f=tid for all 64 rows (coalesced loads)
        const float* Ab = A + (size_t)m0 * CH;
        const int f = tid, kt = f >> 2, ks = f & 3;
        const int laneSlot = 16 * (ks >> 1), sub = ks & 1;
        for (int j = 0; j < 64; ++j) {
            float v = Ab[j * CH + f];
            As[(((kt << 2) + (j >> 4)) << 6) + (((j & 15) + laneSlot) << 1) + sub] = v;
        }
    }
    __syncthreads();
    const int lane = tid & 31;
    const int w    = tid >> 5;
    v8f c[4][2];
    const v8f z = {};
    for (int mt = 0; mt < 4; ++mt) { c[mt][0] = z; c[mt][1] = z; }
    const v2f* wp = Wp + lane;
    for (int kt = 0; kt < 64; ++kt) {
        v2f b0 = wp[(size_t)((kt << 4) + 2 * w) * 32];
        v2f b1 = wp[(size_t)((kt << 4) + 2 * w + 1) * 32];
        const v2f* aF = (const v2f*)&As[kt << 8];
#pragma unroll
        for (int mt = 0; mt < 4; ++mt) {
            v2f a = aF[mt * 32 + lane];
            c[mt][0] = __builtin_amdgcn_wmma_f32_16x16x4_f32(false, a, false, b0,
                                                             (short)0, c[mt][0], false, false);
            c[mt][1] = __builtin_amdgcn_wmma_f32_16x16x4_f32(false, a, false, b1,
                                                             (short)0, c[mt][1], false, false);
        }
    }
    const int colBase = w * 32 + (lane & 15);
    const int rowHi   = (lane >> 4) * 8;
    for (int mt = 0; mt < 4; ++mt) {
#pragma unroll
        for (int v = 0; v < 8; ++v) {
            const int m = m0 + mt * 16 + rowHi + v;
#pragma unroll
            for (int nt = 0; nt < 2; ++nt) {
                const size_t off = (size_t)m * CH + colBase + nt * 16;
                float val = c[mt][nt][v];
                if (Cinit)  val += Cinit[off];
                if (doRelu) val = val > 0.f ? val : 0.f;
                if (Resid)  val += Resid[off];
                C[off] = val;
            }
        }
    }
}

// ---------------------------------------------------------------------------
// Dominant fused kernel: one workgroup per (receiver r, batch b).
//   H[s,:] = relu(ya_t[b,s,:] + yb_t[b,r,:])          (senders s = 0..63)
//   M2     = relu(H @ W2_t^T)                          (64x256 WMMA GEMM)
//   agg[b,r,:] += sum_s rtw[b,r,t,s] * M2[s,:]         (reduced in registers)
// 8192 fp32 WMMA per workgroup; W2 fragments stream from L2.
// ---------------------------------------------------------------------------
__global__ void __launch_bounds__(256)
edge_msg_kernel(const float* __restrict__ ya, const float* __restrict__ yb,
                const v2f* __restrict__ Wp2, const float* __restrict__ rtw,
                float* __restrict__ agg) {
    __shared__ float Hs[16384];                       // one type at a time
    const int r = blockIdx.x, b = blockIdx.y;
    const int tid  = threadIdx.x;
    const int lane = tid & 31;
    const int w    = tid >> 5;
    const int rowHi = (lane >> 4) * 8;
    float acc0 = 0.f, acc1 = 0.f;
    const float* rtwB = rtw + (size_t)(b * NB + r) * 2 * NB;

    for (int type = 0; type < 2; ++type) {
        {   // build H in packed A-fragment order; thread owns column f=tid
            const float* yaB = ya + ((size_t)type * ROWS + b * NB) * CH;
            const float  ybv = yb[((size_t)type * ROWS + b * NB + r) * CH + tid];
            const int f = tid, kt = f >> 2, ks = f & 3;
            const int laneSlot = 16 * (ks >> 1), sub = ks & 1;
            for (int j = 0; j < 64; ++j) {
                float v = yaB[j * CH + f] + ybv;
                v = v > 0.f ? v : 0.f;
                Hs[(((kt << 2) + (j >> 4)) << 6) + (((j & 15) + laneSlot) << 1) + sub] = v;
            }
        }
        __syncthreads();
        const v2f*  wp  = Wp2 + (size_t)type * PKSZ + lane;
        const float* rt = rtwB + type * NB;
        v8f c[4][2];
        const v8f z = {};
        for (int mt = 0; mt < 4; ++mt) { c[mt][0] = z; c[mt][1] = z; }
        for (int kt = 0; kt < 64; ++kt) {
            v2f b0 = wp[(size_t)((kt << 4) + 2 * w) * 32];
            v2f b1 = wp[(size_t)((kt << 4) + 2 * w + 1) * 32];
            const v2f* aF = (const v2f*)&Hs[kt << 8];
#pragma unroll
            for (int mt = 0; mt < 4; ++mt) {
                v2f a = aF[mt * 32 + lane];
                c[mt][0] = __builtin_amdgcn_wmma_f32_16x16x4_f32(false, a, false, b0,
                                                                 (short)0, c[mt][0], false, false);
                c[mt][1] = __builtin_amdgcn_wmma_f32_16x16x4_f32(false, a, false, b1,
                                                                 (short)0, c[mt][1], false, false);
            }
        }
        for (int mt = 0; mt < 4; ++mt) {
#pragma unroll
            for (int v = 0; v < 8; ++v) {
                const float wt = rt[mt * 16 + rowHi + v];
                float e0 = c[mt][0][v]; e0 = e0 > 0.f ? e0 : 0.f;
                float e1 = c[mt][1][v]; e1 = e1 > 0.f ? e1 : 0.f;
                acc0 += wt * e0;
                acc1 += wt * e1;
            }
        }
        __syncthreads();                              // before rebuilding Hs
    }
    // combine the two M-halves (lane L holds rows v, lane L+16 rows v+8)
    acc0 += __shfl_xor(acc0, 16, 32);
    acc1 += __shfl_xor(acc1, 16, 32);
    if (lane < 16) {
        float* out = agg + (size_t)(b * NB + r) * CH + w * 32 + lane;
        out[0]  = acc0;
        out[16] = acc1;
    }
}

// ---------------------------------------------------------------------------
extern "C" void kernel_launch(void* const* d_in, const int* in_sizes, int n_in,
                              void* d_out, int out_size, void* d_ws, size_t ws_size,
                              hipStream_t stream) {
    const float* x        = (const float*)d_in[0];
    const float* rel_type = (const float*)d_in[1];
    const float* w1  = (const float*)d_in[4];
    const float* s1  = (const float*)d_in[5];
    const float* w2  = (const float*)d_in[6];
    const float* s2  = (const float*)d_in[7];
    const float* wo1 = (const float*)d_in[8];
    const float* so1 = (const float*)d_in[9];
    const float* wo2 = (const float*)d_in[10];
    const float* so2 = (const float*)d_in[11];
    float* out = (float*)d_out;

    // workspace layout (floats); total ~13.9 MB
    float* ws     = (float*)d_ws;
    unsigned* thr = (unsigned*)ws;                       //   64 (6 used)
    v2f*   pk     = (v2f*)(ws + 64);                     //   9 * 65536 floats
    float* yaB    = ws + 64 + 9 * 65536;                 //   2 * 2048 * 256
    float* ybB    = yaB + 2 * ROWS * CH;                 //   2 * 2048 * 256
    float* rtw    = ybB + 2 * ROWS * CH;                 //   262144
    float* aggB   = rtw + BT * NB * 2 * NB;              //   524288
    float* tmpB   = yaB;                                 //   reuse: ya dead after edge kernel
    float* pred1  = yaB + ROWS * CH;

    // 1) exact top-50% thresholds per score matrix
    select_thresh_kernel<<<6, 256, 0, stream>>>(s1, s2, so1, so2, thr);
    // 2) mask + pack all 9 weight matrices into WMMA B-fragment layout
    pack_weights_kernel<<<dim3(128, 9), 256, 0, stream>>>(w1, s1, w2, s2,
                                                          wo1, so1, wo2, so2, thr, pk);
    // 3) per-(b,r,type,s) edge-type weights with diagonal zeroed
    build_rtw_kernel<<<1024, 256, 0, stream>>>(rel_type, rtw);
    // 4) layer-1 factored through the one-hot incidence matrices:
    //    ya_t = x @ W1a_t^T (sender half), yb_t = x @ W1b_t^T (receiver half)
    gemm256_wmma_kernel<<<32, 256, 0, stream>>>(x, pk + 0 * PKSZ, nullptr, nullptr, yaB, 0);
    gemm256_wmma_kernel<<<32, 256, 0, stream>>>(x, pk + 1 * PKSZ, nullptr, nullptr, ybB, 0);
    gemm256_wmma_kernel<<<32, 256, 0, stream>>>(x, pk + 2 * PKSZ, nullptr, nullptr, yaB + ROWS * CH, 0);
    gemm256_wmma_kernel<<<32, 256, 0, stream>>>(x, pk + 3 * PKSZ, nullptr, nullptr, ybB + ROWS * CH, 0);
    // 5) fused edge layer-2 + rel_type weighting + receiver scatter (dominant)
    edge_msg_kernel<<<dim3(NB, BT), 256, 0, stream>>>(yaB, ybB, pk + 4 * PKSZ, rtw, aggB);
    // 6) output head: pred = relu([x, agg] @ Wo1^T); out = x + relu(pred @ Wo2^T)
    gemm256_wmma_kernel<<<32, 256, 0, stream>>>(x,     pk + 6 * PKSZ, nullptr, nullptr, tmpB,  0);
    gemm256_wmma_kernel<<<32, 256, 0, stream>>>(aggB,  pk + 7 * PKSZ, tmpB,    nullptr, pred1, 1);
    gemm256_wmma_kernel<<<32, 256, 0, stream>>>(pred1, pk + 8 * PKSZ, nullptr, x,       out,   1);
    (void)in_sizes; (void)n_in; (void)out_size; (void)ws_size;
}